// GDeFeaturizer_23158463660637
// MI455X (gfx1250) — compile-verified
//
#include <hip/hip_runtime.h>

#define NFEAT 64
#define TPTS 4
#define KP 104  // padded Chebyshev order count (multiple of 4 >= 101)

typedef float v2f __attribute__((ext_vector_type(2)));
typedef float v8f __attribute__((ext_vector_type(8)));

// cmat[t*KP + k] = coefsx[t,k] + coefsy[t,k], zero-padded for k >= korders
__global__ void combine_coefs(const float* __restrict__ cx, const float* __restrict__ cy,
                              float* __restrict__ cmat, int korders) {
  int i = blockIdx.x * blockDim.x + threadIdx.x;
  if (i >= TPTS * KP) return;
  int t = i / KP, k = i % KP;
  cmat[i] = (k < korders) ? (cx[t * korders + k] + cy[t * korders + k]) : 0.f;
}

__global__ void count_edges(const int* __restrict__ ei, int* __restrict__ counts, int E) {
  int e = blockIdx.x * blockDim.x + threadIdx.x;
  if (e < E) atomicAdd(&counts[ei[E + e]], 1);
}

// single-workgroup chained exclusive scan over n counts -> row_ptr[0..n]
__global__ void scan_counts(const int* __restrict__ counts, int* __restrict__ row_ptr, int n) {
  __shared__ int sdata[1024];
  __shared__ int srun;
  if (threadIdx.x == 0) srun = 0;
  __syncthreads();
  for (int base = 0; base < n; base += 1024) {
    int i = base + (int)threadIdx.x;
    int v = (i < n) ? counts[i] : 0;
    sdata[threadIdx.x] = v;
    __syncthreads();
    for (int off = 1; off < 1024; off <<= 1) {
      int t = (threadIdx.x >= (unsigned)off) ? sdata[threadIdx.x - off] : 0;
      __syncthreads();
      sdata[threadIdx.x] += t;
      __syncthreads();
    }
    int run = srun;
    if (i < n) row_ptr[i] = run + sdata[threadIdx.x] - v;
    __syncthreads();
    if (threadIdx.x == 1023) srun = run + sdata[1023];
    __syncthreads();
  }
  if (threadIdx.x == 0) row_ptr[n] = srun;
}

__global__ void fill_csr(const int* __restrict__ ei, const float* __restrict__ ew,
                         const int* __restrict__ row_ptr, int* __restrict__ fill,
                         int* __restrict__ ccol, float* __restrict__ cw, int E) {
  int e = blockIdx.x * blockDim.x + threadIdx.x;
  if (e >= E) return;
  int d = ei[E + e];
  int pos = row_ptr[d] + atomicAdd(&fill[d], 1);
  ccol[pos] = ei[e];
  cw[pos] = ew[e];
}

// Fused Chebyshev step: tnew = first ? A*tcur : 2*(A*tcur) - tprev
// One thread per (node, feature-pair): float2 gathers -> global_load_b64, coalesced.
__global__ void cheb_step(const float* __restrict__ tcur, const float* __restrict__ tprev,
                          float* __restrict__ tnew, const int* __restrict__ row_ptr,
                          const int* __restrict__ ccol, const float* __restrict__ cw,
                          int n_nodes, int first) {
  int gid = blockIdx.x * blockDim.x + threadIdx.x;
  int node = gid >> 5;
  if (node >= n_nodes) return;
  int f2 = (gid & 31) * 2;
  int beg = row_ptr[node], end = row_ptr[node + 1];
  float sx = 0.f, sy = 0.f;
  for (int e = beg; e < end; ++e) {
    float wgt = cw[e];
    const float2 hv = *(const float2*)(tcur + ccol[e] * NFEAT + f2);
    sx = fmaf(wgt, hv.x, sx);
    sy = fmaf(wgt, hv.y, sy);
  }
  float2 r;
  if (first) {
    r.x = sx; r.y = sy;
  } else {
    int o = node * NFEAT + f2;
    r.x = fmaf(2.f, sx, -tprev[o]);
    r.y = fmaf(2.f, sy, -tprev[o + 1]);
  }
  *(float2*)(tnew + node * NFEAT + f2) = r;
}

// Group epilogue via V_WMMA_F32_16X16X4_F32:
//   D(16 nf x 16 t) = A(16 nf x 4 k) * B(4 k x 16 t) + C
// A lane layout (32-bit 16x4): VGPR0 = K0(lanes0-15)/K2(lanes16-31), VGPR1 = K1/K3.
// Only t<4 columns are live; B columns 4..15 are zero.
__global__ void accum_group(const float* __restrict__ T0, const float* __restrict__ T1,
                            const float* __restrict__ T2, const float* __restrict__ T3,
                            const float* __restrict__ cmat, float* __restrict__ out,
                            int k0, int n_tiles) {
  int wid = (int)((blockIdx.x * blockDim.x + threadIdx.x) >> 5);
  if (wid >= n_tiles) return;
  int lane = threadIdx.x & 31;
  int hi = lane >> 4;
  int m = lane & 15;              // A row (nf) for A; column t for B/C/D
  int base = wid * 16;            // node-feature base of this tile
  int node = base >> 6;
  int fbase = base & 63;

  const float* Ta = hi ? T2 : T0;
  const float* Tb = hi ? T3 : T1;
  v2f a;
  a.x = Ta[base + m];
  a.y = Tb[base + m];

  int kA = hi ? 2 : 0, kB = hi ? 3 : 1;
  bool act = (m < TPTS);
  v2f b;
  b.x = act ? cmat[m * KP + k0 + kA] : 0.f;
  b.y = act ? cmat[m * KP + k0 + kB] : 0.f;

  // C/D layout: VGPR j -> row M = j + hi*8, col N = m (= t)
  float* obase = out + node * (TPTS * NFEAT) + m * NFEAT + fbase + hi * 8;
  v8f c = {};
  if (act) {
    const float4* p = (const float4*)obase;
    float4 c0 = p[0], c1 = p[1];
    c[0] = c0.x; c[1] = c0.y; c[2] = c0.z; c[3] = c0.w;
    c[4] = c1.x; c[5] = c1.y; c[6] = c1.z; c[7] = c1.w;
  }

  v8f d = __builtin_amdgcn_wmma_f32_16x16x4_f32(false, a, false, b, (short)0, c, false, false);

  if (act) {
    float4* p = (float4*)obase;
    p[0] = make_float4(d[0], d[1], d[2], d[3]);
    p[1] = make_float4(d[4], d[5], d[6], d[7]);
  }
}

extern "C" void kernel_launch(void* const* d_in, const int* in_sizes, int n_in,
                              void* d_out, int out_size, void* d_ws, size_t ws_size,
                              hipStream_t stream) {
  const float* x  = (const float*)d_in[0];
  const int*   ei = (const int*)d_in[1];
  const float* ew = (const float*)d_in[2];
  const float* cx = (const float*)d_in[3];
  const float* cy = (const float*)d_in[4];
  const int N    = in_sizes[0] / NFEAT;   // 50000
  const int E    = in_sizes[2];           // 800000
  const int KORD = in_sizes[3] / TPTS;    // 101 (degree 100)
  const int DEG  = KORD - 1;
  float* out = (float*)d_out;

  // workspace carve (256B aligned): cmat + histogram/CSR + 5-slot T_k ring (~72 MB)
  size_t off = 0;
  auto alloc = [&](size_t bytes) -> void* {
    void* p = (char*)d_ws + off;
    off += (bytes + 255) & ~(size_t)255;
    return p;
  };
  float* cmat   = (float*)alloc((size_t)TPTS * KP * sizeof(float));
  int*   counts = (int*)  alloc((size_t)N * sizeof(int));
  int*   rowptr = (int*)  alloc((size_t)(N + 1) * sizeof(int));
  int*   fill   = (int*)  alloc((size_t)N * sizeof(int));
  int*   ccol   = (int*)  alloc((size_t)E * sizeof(int));
  float* cw     = (float*)alloc((size_t)E * sizeof(float));
  size_t slot = (size_t)N * NFEAT;
  float* ring[5];
  for (int i = 0; i < 5; ++i) ring[i] = (float*)alloc(slot * sizeof(float));
  (void)ws_size;

  hipMemsetAsync(counts, 0, (size_t)N * sizeof(int), stream);
  hipMemsetAsync(fill,   0, (size_t)N * sizeof(int), stream);
  hipMemsetAsync(out,    0, (size_t)out_size * sizeof(float), stream);

  combine_coefs<<<(TPTS * KP + 255) / 256, 256, 0, stream>>>(cx, cy, cmat, KORD);
  count_edges<<<(E + 255) / 256, 256, 0, stream>>>(ei, counts, E);
  scan_counts<<<1, 1024, 0, stream>>>(counts, rowptr, N);
  fill_csr<<<(E + 255) / 256, 256, 0, stream>>>(ei, ew, rowptr, fill, ccol, cw, E);

  // T0 = x
  hipMemcpyAsync(ring[0], x, slot * sizeof(float), hipMemcpyDeviceToDevice, stream);

  const int pthreads = N * 32;
  const int pblocks = (pthreads + 255) / 256;
  const int n_tiles = (int)(slot / 16);
  const int ablocks = (n_tiles * 32 + 255) / 256;

  // T1 = A*T0
  cheb_step<<<pblocks, 256, 0, stream>>>(ring[0], ring[0], ring[1], rowptr, ccol, cw, N, 1);

  for (int k = 2; k <= DEG; ++k) {
    cheb_step<<<pblocks, 256, 0, stream>>>(ring[(k - 1) % 5], ring[(k - 2) % 5], ring[k % 5],
                                           rowptr, ccol, cw, N, 0);
    if ((k & 3) == 3) {
      int k0 = k - 3;
      accum_group<<<ablocks, 256, 0, stream>>>(ring[k0 % 5], ring[(k0 + 1) % 5],
                                               ring[(k0 + 2) % 5], ring[(k0 + 3) % 5],
                                               cmat, out, k0, n_tiles);
    }
  }
  // tail group (orders past the last full 4-group; padded coefs are zero)
  if ((DEG & 3) != 3) {
    int k0 = (DEG / 4) * 4;
    const float* t0 = ring[(k0 + 0 > DEG ? DEG : k0 + 0) % 5];
    const float* t1 = ring[(k0 + 1 > DEG ? DEG : k0 + 1) % 5];
    const float* t2 = ring[(k0 + 2 > DEG ? DEG : k0 + 2) % 5];
    const float* t3 = ring[(k0 + 3 > DEG ? DEG : k0 + 3) % 5];
    accum_group<<<ablocks, 256, 0, stream>>>(t0, t1, t2, t3, cmat, out, k0, n_tiles);
  }
}